// SimpleRandomForest_5488968204626
// MI455X (gfx1250) — compile-verified
//
#include <hip/hip_runtime.h>
#include <hip/hip_bf16.h>

typedef __attribute__((ext_vector_type(2))) float v2f;
typedef __attribute__((ext_vector_type(8))) float v8f;
typedef __attribute__((ext_vector_type(4))) int   v4i;
typedef __attribute__((address_space(1))) v4i*    gp4;   // global int4*
typedef __attribute__((address_space(3))) v4i*    lp4;   // LDS int4*

#define BB 65536
#define FF 1024
#define TT 64
#define DD 6
#define CC 384          // T*D columns of the GEMM
#define MB 64           // batch rows per workgroup
#define KC 64           // K chunk staged in LDS
#define PITCH 68        // LDS row pitch in floats: rows 16B-aligned, banks spread
#define NJ 6            // N-tiles per wave (24 tiles / 4 wave columns)
#define NI 2            // M-tiles per wave (4 tiles / 2 wave rows)

// ---- CDNA5 async global->LDS staging (ASYNCcnt) with safe fallback ----
#define USE_ASYNC 0
#if defined(__has_builtin)
#if __has_builtin(__builtin_amdgcn_global_load_async_to_lds_b128) && \
    __has_builtin(__builtin_amdgcn_s_wait_asynccnt)
#undef USE_ASYNC
#define USE_ASYNC 1
#endif
#endif

__device__ __forceinline__ void stage16(const float* __restrict__ src, float* dst) {
#if USE_ASYNC
    float* s = const_cast<float*>(src);
    __builtin_amdgcn_global_load_async_to_lds_b128(
        (gp4)(v4i*)s, (lp4)(v4i*)dst, /*offset=*/0, /*cpol=*/0);
#else
    *(float4*)dst = *(const float4*)src;
#endif
}

__device__ __forceinline__ void stage_wait() {
#if USE_ASYNC
    __builtin_amdgcn_s_wait_asynccnt(0);
#endif
}

__global__ __launch_bounds__(256) void forest_wmma_kernel(
    const float* __restrict__ x,          // (B, F)
    const float* __restrict__ splits,     // (T, D, F) -> col c = t*6+d contiguous in F
    const float* __restrict__ thresholds, // (T, D) flat = per-column threshold
    const float* __restrict__ values,     // (T, 64)
    float* __restrict__ out)              // (B, 1)
{
    extern __shared__ float smem[];
    float* lx = smem;                                   // MB * PITCH
    float* ls = smem + MB * PITCH;                      // CC * PITCH
    unsigned int* code = (unsigned int*)(smem + (MB + CC) * PITCH); // MB*TT

    const int tid   = threadIdx.x;
    const int lane  = tid & 31;
    const int wave  = tid >> 5;
    const int wm    = wave & 1;          // wave row: M-tiles {2*wm, 2*wm+1}
    const int wn    = wave >> 1;         // wave col: N-tiles wn + 4*j
    const int gm    = blockIdx.x * MB;   // global row base

    const int lm    = lane & 15;         // fragment row/col within tile
    const int khalf = (lane >> 4) << 1;  // 0 or 2 : K sub-offset per ISA layout

    v8f acc[NI][NJ];
    #pragma unroll
    for (int i = 0; i < NI; ++i)
        #pragma unroll
        for (int j = 0; j < NJ; ++j) acc[i][j] = {};

    for (int kc = 0; kc < FF / KC; ++kc) {
        const int kbase = kc * KC;
        // ---- stage x tile: 64 rows x 64 floats (coalesced b128, async->LDS) ----
        #pragma unroll
        for (int i = 0; i < (MB * KC / 4) / 256; ++i) {  // 4 iters
            int q = tid + i * 256;                       // float4 index
            int r = q >> 4;
            int c = (q & 15) << 2;
            stage16(&x[(size_t)(gm + r) * FF + kbase + c], &lx[r * PITCH + c]);
        }
        // ---- stage splits tile: 384 rows x 64 floats ----
        #pragma unroll
        for (int i = 0; i < (CC * KC / 4) / 256; ++i) {  // 24 iters
            int q = tid + i * 256;
            int r = q >> 4;
            int c = (q & 15) << 2;
            stage16(&splits[(size_t)r * FF + kbase + c], &ls[r * PITCH + c]);
        }
        stage_wait();
        __syncthreads();

        // ---- 16 WMMA K-steps of 4 over this chunk ----
        #pragma unroll 4
        for (int kk = 0; kk < KC / 4; ++kk) {
            const int kb = (kk << 2) + khalf;
            v2f a[NI];
            #pragma unroll
            for (int i = 0; i < NI; ++i) {
                const int mt = wm * 2 + i;
                a[i] = *(const v2f*)&lx[(mt * 16 + lm) * PITCH + kb];
            }
            #pragma unroll
            for (int j = 0; j < NJ; ++j) {
                const int nt = wn + 4 * j;
                v2f b = *(const v2f*)&ls[(nt * 16 + lm) * PITCH + kb];
                #pragma unroll
                for (int i = 0; i < NI; ++i) {
                    acc[i][j] = __builtin_amdgcn_wmma_f32_16x16x4_f32(
                        false, a[i], false, b, (short)0, acc[i][j], false, false);
                }
            }
        }
        __syncthreads();
    }

    // ---- phase 2: threshold -> pack decision bits into per-(row,tree) codes ----
    #pragma unroll
    for (int i = 0; i < (MB * TT) / 256; ++i) code[tid + i * 256] = 0u;
    __syncthreads();

    const int vbase = (lane >> 4) << 3;             // acc element v lives at row vbase+v
    #pragma unroll
    for (int j = 0; j < NJ; ++j) {
        const int col = (wn + 4 * j) * 16 + lm;     // 0..383 == t*6+d
        const int t = col / DD;
        const int d = col - t * DD;
        const float thr = thresholds[col];
        const unsigned int mask = 1u << (5 - d);
        #pragma unroll
        for (int i = 0; i < NI; ++i) {
            const int rbase = (wm * 2 + i) * 16 + vbase;
            #pragma unroll
            for (int v = 0; v < 8; ++v) {
                if (acc[i][j][v] > thr)
                    atomicOr(&code[(rbase + v) * TT + t], mask); // ds_or_b32
            }
        }
    }
    __syncthreads();

    // ---- phase 3: leaf gather + mean over trees ----
    if (tid < MB) {
        float s = 0.0f;
        #pragma unroll 8
        for (int t = 0; t < TT; ++t) {
            unsigned int c = code[tid * TT + t];
            int leaf = (int)((c - 1u) & 63u);       // (code-1) mod 64
            s += values[t * 64 + leaf];
        }
        out[gm + tid] = s * (1.0f / TT);
    }
}

extern "C" void kernel_launch(void* const* d_in, const int* in_sizes, int n_in,
                              void* d_out, int out_size, void* d_ws, size_t ws_size,
                              hipStream_t stream) {
    const float* x          = (const float*)d_in[0];
    const float* splits     = (const float*)d_in[1];
    const float* thresholds = (const float*)d_in[2];
    const float* values     = (const float*)d_in[3];
    float* out = (float*)d_out;

    size_t shmem = (size_t)(MB + CC) * PITCH * sizeof(float)   // staging tiles
                 + (size_t)MB * TT * sizeof(unsigned int);     // code array  (~135 KB)
    dim3 grid(BB / MB), block(256);
    forest_wmma_kernel<<<grid, block, shmem, stream>>>(x, splits, thresholds, values, out);
}